// AggrGSMean_19645180412609
// MI455X (gfx1250) — compile-verified
//
#include <hip/hip_runtime.h>
#include <stdint.h>
#include <stddef.h>

// Problem constants (from the reference): B=1, V=100000, T=2, N=32, F=64, M=1600000
#define Vv   100000
#define Tt   2
#define Nn   32
#define Ff   64
#define Mm   1600000
#define NSEG (Tt * Vv)                      // 200000 segments (t-major: seg = t*V + v)
#define CHUNK 1024
#define NCH  ((NSEG + CHUNK - 1) / CHUNK)   // 196 scan chunks (<= 256 for single-block base scan)

#if __has_builtin(__builtin_amdgcn_global_load_async_to_lds_b32) && \
    __has_builtin(__builtin_amdgcn_s_wait_asynccnt)
#define HAVE_ASYNC_LDS 1
#endif

// ---------------------------------------------------------------------------
// K1: zero the per-segment counters (workspace is poisoned / left dirty).
// ---------------------------------------------------------------------------
__global__ __launch_bounds__(256) void k_zero(int* __restrict__ p, int n) {
  int i = blockIdx.x * 256 + threadIdx.x;
  if (i < n) p[i] = 0;
}

// ---------------------------------------------------------------------------
// K2: histogram — counts[t*V + v] += 1 for every row m (integer atomics only).
// ---------------------------------------------------------------------------
__global__ __launch_bounds__(256) void k_count(const int* __restrict__ idx0,
                                               const int* __restrict__ idx1,
                                               int* __restrict__ cnt) {
  int m = blockIdx.x * 256 + threadIdx.x;
  if (m >= Mm) return;
  int t = blockIdx.y;
  const int* __restrict__ idx = t ? idx1 : idx0;
  int v = idx[m * 3 + 1];
  atomicAdd(&cnt[t * Vv + v], 1);
}

// ---------------------------------------------------------------------------
// K3a: per-chunk (1024 elems) sums of the counts array.
// ---------------------------------------------------------------------------
__global__ __launch_bounds__(256) void k_chunk_sum(const int* __restrict__ cnt,
                                                   int* __restrict__ bsum, int n) {
  __shared__ int lds[256];
  int tid = threadIdx.x;
  int base = blockIdx.x * CHUNK + tid * 4;
  int s = 0;
  for (int j = 0; j < 4; ++j) {
    int gi = base + j;
    if (gi < n) s += cnt[gi];
  }
  lds[tid] = s;
  __syncthreads();
  for (int off = 128; off > 0; off >>= 1) {
    if (tid < off) lds[tid] += lds[tid + off];
    __syncthreads();
  }
  if (tid == 0) bsum[blockIdx.x] = lds[0];
}

// ---------------------------------------------------------------------------
// K3b: single-block exclusive scan of the (<=256) chunk sums, in place.
// ---------------------------------------------------------------------------
__global__ __launch_bounds__(256) void k_scan_bases(int* __restrict__ bsum, int nchunks) {
  __shared__ int lds[256];
  int tid = threadIdx.x;
  int val = (tid < nchunks) ? bsum[tid] : 0;
  lds[tid] = val;
  __syncthreads();
  for (int off = 1; off < 256; off <<= 1) {  // Hillis-Steele inclusive scan
    int add = (tid >= off) ? lds[tid - off] : 0;
    __syncthreads();
    lds[tid] += add;
    __syncthreads();
  }
  if (tid < nchunks) bsum[tid] = (tid == 0) ? 0 : lds[tid - 1];
}

// ---------------------------------------------------------------------------
// K3c: local exclusive scan of each 1024-chunk + chunk base -> offsets (in place).
// ---------------------------------------------------------------------------
__global__ __launch_bounds__(256) void k_scan_local(int* __restrict__ data,
                                                    const int* __restrict__ bases, int n) {
  __shared__ int lds[256];
  int tid = threadIdx.x;
  int base = blockIdx.x * CHUNK + tid * 4;
  int x[4];
  int s = 0;
  for (int j = 0; j < 4; ++j) {
    int gi = base + j;
    x[j] = (gi < n) ? data[gi] : 0;
    s += x[j];
  }
  lds[tid] = s;
  __syncthreads();
  for (int off = 1; off < 256; off <<= 1) {
    int add = (tid >= off) ? lds[tid - off] : 0;
    __syncthreads();
    lds[tid] += add;
    __syncthreads();
  }
  int run = bases[blockIdx.x] + ((tid == 0) ? 0 : lds[tid - 1]);
  for (int j = 0; j < 4; ++j) {
    int gi = base + j;
    if (gi < n) data[gi] = run;
    run += x[j];
  }
}

// ---------------------------------------------------------------------------
// K4: fill buckets. After this, offs[seg] holds the segment END (== next start).
// ---------------------------------------------------------------------------
__global__ __launch_bounds__(256) void k_fill(const int* __restrict__ idx0,
                                              const int* __restrict__ idx1,
                                              int* __restrict__ offs,
                                              int* __restrict__ bucket) {
  int m = blockIdx.x * 256 + threadIdx.x;
  if (m >= Mm) return;
  int t = blockIdx.y;
  const int* __restrict__ idx = t ? idx1 : idx0;
  int v = idx[m * 3 + 1];
  int pos = atomicAdd(&offs[t * Vv + v], 1);
  bucket[pos] = m;
}

// ---------------------------------------------------------------------------
// K5: gather. One 64-thread block per segment (t,v); thread f owns column f.
//  - bucket slice pulled into LDS with CDNA5 async global->LDS copy (ASYNCcnt)
//  - rank-sorted ascending-m (deterministic across graph replays)
//  - 8 independent accumulators -> 8 outstanding coalesced 128B loads per wave
//    (Little's law: need >10MB in flight fleet-wide to reach 23.3 TB/s)
//  - degree via wave32 ballot: lane n tests adjacency slot n (N == wave32 == 32)
// ---------------------------------------------------------------------------
__global__ __launch_bounds__(Ff) void k_gather(const int* __restrict__ adj,
                                               const float* __restrict__ feat0,
                                               const float* __restrict__ feat1,
                                               const int* __restrict__ offs,
                                               const int* __restrict__ bucket,
                                               float* __restrict__ out) {
  const int i = blockIdx.x;                 // segment = t*V + v
  const int f = threadIdx.x;                // 0..63
  const int start = (i == 0) ? 0 : offs[i - 1];
  const int end = offs[i];
  const int cnt = end - start;
  const int t = (i >= Vv) ? 1 : 0;
  const int v = i - t * Vv;
  const float* __restrict__ feat = t ? feat1 : feat0;
  const float* __restrict__ col = feat + f;   // this thread's feature column

  __shared__ int ent[Ff];
  __shared__ int srt[Ff];

  float p0 = 0.f, p1 = 0.f, p2 = 0.f, p3 = 0.f;
  float p4 = 0.f, p5 = 0.f, p6 = 0.f, p7 = 0.f;

  if (cnt <= Ff) {  // statistically always true (Poisson mean 16)
#if defined(HAVE_ASYNC_LDS)
    if (f < cnt)
      __builtin_amdgcn_global_load_async_to_lds_b32(
          (int*)(bucket + start + f), (int*)&ent[f], 0, 0);
    __builtin_amdgcn_s_wait_asynccnt(0);
#else
    if (f < cnt) ent[f] = bucket[start + f];
#endif
    __syncthreads();
    if (f < cnt) {  // rank sort: m values are distinct within a segment
      const int me = ent[f];
      int r = 0;
      for (int j = 0; j < cnt; ++j) r += (ent[j] < me) ? 1 : 0;
      srt[r] = me;
    }
    __syncthreads();
    int e = 0;
    for (; e + 8 <= cnt; e += 8) {  // 8 loads in flight per wave
      p0 += __builtin_nontemporal_load(col + (size_t)srt[e + 0] * Ff);
      p1 += __builtin_nontemporal_load(col + (size_t)srt[e + 1] * Ff);
      p2 += __builtin_nontemporal_load(col + (size_t)srt[e + 2] * Ff);
      p3 += __builtin_nontemporal_load(col + (size_t)srt[e + 3] * Ff);
      p4 += __builtin_nontemporal_load(col + (size_t)srt[e + 4] * Ff);
      p5 += __builtin_nontemporal_load(col + (size_t)srt[e + 5] * Ff);
      p6 += __builtin_nontemporal_load(col + (size_t)srt[e + 6] * Ff);
      p7 += __builtin_nontemporal_load(col + (size_t)srt[e + 7] * Ff);
    }
    for (; e < cnt; ++e)
      p0 += __builtin_nontemporal_load(col + (size_t)srt[e] * Ff);
  } else {  // overflow fallback (unsorted)
    int e = start;
    for (; e + 8 <= end; e += 8) {
      p0 += __builtin_nontemporal_load(col + (size_t)bucket[e + 0] * Ff);
      p1 += __builtin_nontemporal_load(col + (size_t)bucket[e + 1] * Ff);
      p2 += __builtin_nontemporal_load(col + (size_t)bucket[e + 2] * Ff);
      p3 += __builtin_nontemporal_load(col + (size_t)bucket[e + 3] * Ff);
      p4 += __builtin_nontemporal_load(col + (size_t)bucket[e + 4] * Ff);
      p5 += __builtin_nontemporal_load(col + (size_t)bucket[e + 5] * Ff);
      p6 += __builtin_nontemporal_load(col + (size_t)bucket[e + 6] * Ff);
      p7 += __builtin_nontemporal_load(col + (size_t)bucket[e + 7] * Ff);
    }
    for (; e < end; ++e)
      p0 += __builtin_nontemporal_load(col + (size_t)bucket[e] * Ff);
  }
  const float acc = ((p0 + p1) + (p2 + p3)) + ((p4 + p5) + (p6 + p7));

  // degree[v,t] = max(#(adj[v,t,n] >= 0), 1); lane n of each wave tests slot n
  int deg;
#if __has_builtin(__builtin_amdgcn_ballot_w32)
  {
    const int lane = f & 31;
    const int a = adj[v * (Tt * Nn) + t * Nn + lane];
    deg = __popc(__builtin_amdgcn_ballot_w32(a >= 0));
  }
#else
  {
    const int* __restrict__ row = adj + v * (Tt * Nn) + t * Nn;
    deg = 0;
    for (int n2 = 0; n2 < Nn; ++n2) deg += (row[n2] >= 0) ? 1 : 0;
  }
#endif
  if (deg < 1) deg = 1;

  __builtin_nontemporal_store(acc / (float)deg, out + (size_t)i * Ff + f);
}

// ---------------------------------------------------------------------------
// Host side. Inputs (setup_inputs order):
//   d_in[0] adjacency int32 [1,V,2,32]
//   d_in[1] indices0  int32 [M,3]
//   d_in[2] features0 f32   [M,64]
//   d_in[3] indices1  int32 [M,3]
//   d_in[4] features1 f32   [M,64]
// d_out: out0 ++ out1, 2*V*64 f32.
// Workspace layout (needs ~13.7 MB):
//   offs   : NSEG ints        (counters -> offsets -> segment ends)
//   bsum   : 1024 ints (pad)  (chunk sums / bases)
//   bucket : 2*M ints         (row ids grouped by segment)
// ---------------------------------------------------------------------------
extern "C" void kernel_launch(void* const* d_in, const int* in_sizes, int n_in,
                              void* d_out, int out_size, void* d_ws, size_t ws_size,
                              hipStream_t stream) {
  (void)in_sizes; (void)n_in; (void)out_size; (void)ws_size;
  const int*   adj  = (const int*)d_in[0];
  const int*   idx0 = (const int*)d_in[1];
  const float* f0   = (const float*)d_in[2];
  const int*   idx1 = (const int*)d_in[3];
  const float* f1   = (const float*)d_in[4];
  float* out = (float*)d_out;

  int* offs   = (int*)d_ws;
  int* bsum   = offs + NSEG;     // 200000
  int* bucket = bsum + 1024;     // padded

  k_zero<<<(NSEG + 255) / 256, 256, 0, stream>>>(offs, NSEG);

  dim3 gM(Mm / 256, Tt);         // 1600000 / 256 == 6250 exactly
  k_count<<<gM, 256, 0, stream>>>(idx0, idx1, offs);

  k_chunk_sum<<<NCH, 256, 0, stream>>>(offs, bsum, NSEG);
  k_scan_bases<<<1, 256, 0, stream>>>(bsum, NCH);
  k_scan_local<<<NCH, 256, 0, stream>>>(offs, bsum, NSEG);

  k_fill<<<gM, 256, 0, stream>>>(idx0, idx1, offs, bucket);

  k_gather<<<NSEG, Ff, 0, stream>>>(adj, f0, f1, offs, bucket, out);
}